// ConvSoftArgmax3d_82738249990630
// MI455X (gfx1250) — compile-verified
//
#include <hip/hip_runtime.h>
#include <hip/hip_bf16.h>
#include <cstdint>

// Problem geometry (fixed by the reference)
#define B_    4
#define C_    8
#define D_    16
#define H_    256
#define W_    256
#define CH_   (B_ * C_)
#define HW_   (H_ * W_)
#define DHW_  (D_ * HW_)
#define EPSV  1e-8f

// Tile: 32(W) x 8(H) x 4(D) outputs per block, +1 halo each side
#define TW   32
#define TH_  8
#define TD   4
#define LW   (TW + 2)   // 34
#define LH   (TH_ + 2)  // 10
#define LD   (TD + 2)   // 6
#define LVOL (LD * LH * LW)  // 2040

#define BPC  64  // blocks per channel in the max pass

// ---- order-preserving float<->uint mapping for atomicMax on floats ----
__device__ __forceinline__ unsigned ordf(float f) {
  unsigned b = __float_as_uint(f);
  return b ^ ((unsigned)((int)b >> 31) | 0x80000000u);
}
__device__ __forceinline__ float unordf(unsigned u) {
  unsigned b = (u & 0x80000000u) ? (u ^ 0x80000000u) : ~u;
  return __uint_as_float(b);
}

// ---- CDNA5 async global -> LDS copy (ASYNCcnt-tracked) ----
__device__ __forceinline__ void async_g2lds_b32(uint32_t lds_byte_addr, uint64_t gaddr) {
  asm volatile("global_load_async_to_lds_b32 %0, %1, off"
               :: "v"(lds_byte_addr), "v"(gaddr)
               : "memory");
}
__device__ __forceinline__ void wait_asynccnt0() {
#if __has_builtin(__builtin_amdgcn_s_wait_asynccnt)
  __builtin_amdgcn_s_wait_asynccnt(0);
#else
  asm volatile("s_wait_asynccnt 0" ::: "memory");
#endif
}

// ============ Pass 0: zero the per-channel max slots ============
__global__ __launch_bounds__(64) void csam_init(unsigned* __restrict__ m) {
  if (threadIdx.x < CH_) m[threadIdx.x] = 0u;
}

// ============ Pass 1: per-channel max ============
__global__ __launch_bounds__(256) void csam_max(const float* __restrict__ x,
                                                unsigned* __restrict__ m) {
  const int chan = blockIdx.x / BPC;
  const int sub  = blockIdx.x % BPC;
  const int per  = DHW_ / BPC;  // 16384
  const float* p = x + (size_t)chan * DHW_ + (size_t)sub * per;

  float lm = -__builtin_inff();
  for (int i = threadIdx.x; i < per; i += 256) lm = fmaxf(lm, p[i]);

  // wave32 reduction
  #pragma unroll
  for (int off = 16; off > 0; off >>= 1)
    lm = fmaxf(lm, __shfl_xor(lm, off, 32));

  __shared__ float wmax[8];
  if ((threadIdx.x & 31) == 0) wmax[threadIdx.x >> 5] = lm;
  __syncthreads();
  if (threadIdx.x == 0) {
    float bm = wmax[0];
    #pragma unroll
    for (int i = 1; i < 8; ++i) bm = fmaxf(bm, wmax[i]);
    atomicMax(m + chan, ordf(bm));
  }
}

// ============ Pass 2: soft-argmax stencil ============
__global__ __launch_bounds__(256) void csam_main(const float* __restrict__ x,
                                                 const unsigned* __restrict__ chmax,
                                                 float* __restrict__ out) {
  __shared__ float xs[LVOL];
  __shared__ float es[LVOL];
  __shared__ float fs[LVOL];

  unsigned bid = blockIdx.x;
  const int tw = bid & 7;   bid >>= 3;   // W/TW  = 8
  const int th = bid & 31;  bid >>= 5;   // H/TH  = 32
  const int td = bid & 3;   bid >>= 2;   // D/TD  = 4
  const int chan = (int)bid;             // 32
  const int w0 = tw * TW, h0 = th * TH_, d0 = td * TD;

  const float* xc = x + (size_t)chan * DHW_;
  const int tid = threadIdx.x;

  // ---- stage halo tile: global -> LDS asynchronously; zero-fill OOB ----
  for (int i = tid; i < LVOL; i += 256) {
    const int lz = i / (LH * LW);
    const int r  = i - lz * (LH * LW);
    const int ly = r / LW;
    const int lx = r - ly * LW;
    const int d = d0 + lz - 1, h = h0 + ly - 1, w = w0 + lx - 1;
    const bool ok = ((unsigned)d < (unsigned)D_) & ((unsigned)h < (unsigned)H_) &
                    ((unsigned)w < (unsigned)W_);
    if (ok) {
      const uint32_t la = (uint32_t)(uintptr_t)(&xs[i]);  // AS3 offset survives truncation
      const uint64_t ga = (uint64_t)(uintptr_t)(xc + ((size_t)d * H_ + h) * W_ + w);
      async_g2lds_b32(la, ga);
    } else {
      xs[i] = 0.0f;
    }
  }
  wait_asynccnt0();
  __syncthreads();

  const float mx = unordf(chmax[chan]);

  // ---- e = exp(x - mx), f = e*x (zero on padded cells) ----
  for (int i = tid; i < LVOL; i += 256) {
    const int lz = i / (LH * LW);
    const int r  = i - lz * (LH * LW);
    const int ly = r / LW;
    const int lx = r - ly * LW;
    const int d = d0 + lz - 1, h = h0 + ly - 1, w = w0 + lx - 1;
    const bool ok = ((unsigned)d < (unsigned)D_) & ((unsigned)h < (unsigned)H_) &
                    ((unsigned)w < (unsigned)W_);
    const float v = xs[i];
    const float e = ok ? expf(v - mx) : 0.0f;
    es[i] = e;
    fs[i] = e * v;
  }
  __syncthreads();

  // ---- each thread owns one (oy,ox) column, all TD output z's ----
  const int ox = tid & 31;
  const int oy = tid >> 5;

  float p1[LD], pxw[LD], pyw[LD], pf[LD];
  #pragma unroll
  for (int lz = 0; lz < LD; ++lz) {
    const int base = (lz * LH + oy) * LW + ox;
    const float e00 = es[base],          e01 = es[base + 1],          e02 = es[base + 2];
    const float e10 = es[base + LW],     e11 = es[base + LW + 1],     e12 = es[base + LW + 2];
    const float e20 = es[base + 2 * LW], e21 = es[base + 2 * LW + 1], e22 = es[base + 2 * LW + 2];
    const float r0 = e00 + e01 + e02;
    const float r1 = e10 + e11 + e12;
    const float r2 = e20 + e21 + e22;
    p1[lz]  = r0 + r1 + r2;
    pyw[lz] = r2 - r0;                              // dy in {-1,0,+1}
    pxw[lz] = (e02 + e12 + e22) - (e00 + e10 + e20); // dx in {-1,0,+1}
    pf[lz]  = fs[base]          + fs[base + 1]          + fs[base + 2]
            + fs[base + LW]     + fs[base + LW + 1]     + fs[base + LW + 2]
            + fs[base + 2 * LW] + fs[base + 2 * LW + 1] + fs[base + 2 * LW + 2];
  }

  float* oc = out + (size_t)chan * 3 * DHW_;                         // coords, ch-major
  float* ov = out + (size_t)CH_ * 3 * DHW_ + (size_t)chan * DHW_;    // vals
  const float fw = (float)(w0 + ox);
  const float fh = (float)(h0 + oy);

  #pragma unroll
  for (int oz = 0; oz < TD; ++oz) {
    const float den  = p1[oz] + p1[oz + 1] + p1[oz + 2] + EPSV;
    const float rden = 1.0f / den;
    const float cz = (p1[oz + 2] - p1[oz]) * rden + (float)(d0 + oz);
    const float cx = (pxw[oz] + pxw[oz + 1] + pxw[oz + 2]) * rden + fw;
    const float cy = (pyw[oz] + pyw[oz + 1] + pyw[oz + 2]) * rden + fh;
    const float vv = (pf[oz] + pf[oz + 1] + pf[oz + 2]) * rden;

    const size_t sp = (size_t)(d0 + oz) * HW_ + (size_t)(h0 + oy) * W_ + (size_t)(w0 + ox);
    __builtin_nontemporal_store(cz, oc + sp);                       // channel z
    __builtin_nontemporal_store(cx, oc + (size_t)DHW_ + sp);        // channel x
    __builtin_nontemporal_store(cy, oc + 2 * (size_t)DHW_ + sp);    // channel y
    __builtin_nontemporal_store(vv, ov + sp);                       // vals
  }
}

extern "C" void kernel_launch(void* const* d_in, const int* in_sizes, int n_in,
                              void* d_out, int out_size, void* d_ws, size_t ws_size,
                              hipStream_t stream) {
  (void)in_sizes; (void)n_in; (void)out_size; (void)ws_size;
  const float* x = (const float*)d_in[0];
  float* out = (float*)d_out;
  unsigned* chmax = (unsigned*)d_ws;  // 32 slots

  csam_init<<<1, 64, 0, stream>>>(chmax);
  csam_max<<<CH_ * BPC, 256, 0, stream>>>(x, chmax);

  const int nblocks = CH_ * (D_ / TD) * (H_ / TH_) * (W_ / TW);  // 32768
  csam_main<<<nblocks, 256, 0, stream>>>(x, chmax, out);
}